// FPLoss_43155831390667
// MI455X (gfx1250) — compile-verified
//
#include <hip/hip_runtime.h>
#include <math.h>

#define IMG    416
#define N_ANG  320
#define N_DET  416
#define N_SAMP 416
#define N_RAYS (N_ANG * N_DET)
#define DIST_F 1075.0f
#define CTR_F  207.5f

typedef float v2f __attribute__((ext_vector_type(2)));
typedef float v8f __attribute__((ext_vector_type(8)));

// ---------------------------------------------------------------------------
// Wave32 sum reduction using V_WMMA_F32_16X16X4_F32.
// A (16x4 f32, 2 VGPRs): VGPR0 = per-lane value v, VGPR1 = 0
//   -> A[m][0] = v[m] (lanes 0-15), A[m][2] = v[m+16] (lanes 16-31)
// B (4x16 f32, 2 VGPRs) = all ones
//   -> D[m][n] = v[m] + v[m+16]
// Summing the 8 D VGPRs in a lane gives:
//   lanes 0-15 : sum(v[0..7])  + sum(v[16..23])
//   lanes 16-31: sum(v[8..15]) + sum(v[24..31])
// One shfl_xor(16) completes the full 32-lane sum (exact f32 adds).
// Requires EXEC == all ones (true: full 256-thread blocks, no divergence).
// ---------------------------------------------------------------------------
__device__ inline float wave_sum_wmma(float v) {
    v2f a; a[0] = v;    a[1] = 0.0f;
    v2f b; b[0] = 1.0f; b[1] = 1.0f;
    v8f c = {};
    c = __builtin_amdgcn_wmma_f32_16x16x4_f32(
            /*neg_a=*/false, a, /*neg_b=*/false, b,
            /*c_mod=*/(short)0, c, /*reuse_a=*/false, /*reuse_b=*/false);
    float s = ((c[0] + c[1]) + (c[2] + c[3])) + ((c[4] + c[5]) + (c[6] + c[7]));
    s += __shfl_xor(s, 16, 32);
    return s;
}

// ---------------------------------------------------------------------------
// D = input - target, and zero the scalar accumulator.
// ---------------------------------------------------------------------------
__global__ __launch_bounds__(256) void fb_diff_kernel(const float* __restrict__ a,
                                                      const float* __restrict__ b,
                                                      float* __restrict__ d,
                                                      float* __restrict__ acc) {
    int i = blockIdx.x * 256 + threadIdx.x;
    if (i < IMG * IMG) d[i] = a[i] - b[i];
    if (i == 0) *acc = 0.0f;
}

__global__ void fb_zero_acc(float* acc) { *acc = 0.0f; }

// ---------------------------------------------------------------------------
// One ray per lane. DIFF=true: sample precomputed diff image (4 loads/sample).
// DIFF=false: sample both images (8 loads/sample), subtract per texel.
// ---------------------------------------------------------------------------
template <bool DIFF>
__global__ __launch_bounds__(256) void fb_ray_kernel(const float* __restrict__ imgA,
                                                     const float* __restrict__ imgB,
                                                     float* __restrict__ acc,
                                                     float gmax, float step_gm,
                                                     float step_th, float t0, float dt) {
    const int ray = blockIdx.x * 256 + threadIdx.x;
    const int det = ray % N_DET;
    const int ang = ray / N_DET;

    const float th = (float)ang * step_th;
    const float gm = fmaf((float)det, step_gm, -gmax);
    float sth, cth, sg, cg;
    __sincosf(th, &sth, &cth);
    __sincosf(th + gm, &sg, &cg);

    const float x0 = fmaf(-t0, cg, DIST_F * cth) + CTR_F;
    const float y0 = fmaf(-t0, sg, DIST_F * sth) + CTR_F;
    const float dx = -dt * cg;
    const float dy = -dt * sg;

    float S = 0.0f;
    #pragma unroll 4
    for (int k = 0; k < N_SAMP; ++k) {
        const float x = fmaf((float)k, dx, x0);
        const float y = fmaf((float)k, dy, y0);
        const float cf = floorf(x), rf = floorf(y);
        const float wc = x - cf,   wr = y - rf;
        const int c0 = (int)cf, r0 = (int)rf;
        const int c1 = c0 + 1,  r1 = r0 + 1;

        const int cc0 = min(max(c0, 0), IMG - 1);
        const int cc1 = min(max(c1, 0), IMG - 1);
        const int rr0 = min(max(r0, 0), IMG - 1) * IMG;
        const int rr1 = min(max(r1, 0), IMG - 1) * IMG;

        float v00 = imgA[rr0 + cc0];
        float v01 = imgA[rr0 + cc1];
        float v10 = imgA[rr1 + cc0];
        float v11 = imgA[rr1 + cc1];
        if (!DIFF) {
            v00 -= imgB[rr0 + cc0];
            v01 -= imgB[rr0 + cc1];
            v10 -= imgB[rr1 + cc0];
            v11 -= imgB[rr1 + cc1];
        }

        const bool rin0 = (unsigned)r0 < (unsigned)IMG;
        const bool rin1 = (unsigned)r1 < (unsigned)IMG;
        const bool cin0 = (unsigned)c0 < (unsigned)IMG;
        const bool cin1 = (unsigned)c1 < (unsigned)IMG;
        v00 = (rin0 && cin0) ? v00 : 0.0f;
        v01 = (rin0 && cin1) ? v01 : 0.0f;
        v10 = (rin1 && cin0) ? v10 : 0.0f;
        v11 = (rin1 && cin1) ? v11 : 0.0f;

        const float top = fmaf(wc, v01 - v00, v00);
        const float bot = fmaf(wc, v11 - v10, v10);
        S += fmaf(wr, bot - top, top);
    }

    // per-ray contribution (dt^2 folded into final scale)
    const float wsum = wave_sum_wmma(S * S);
    if ((threadIdx.x & 31u) == 0u) {
        unsafeAtomicAdd(acc, wsum);
    }
}

__global__ void fb_finalize(const float* __restrict__ acc, float* __restrict__ out,
                            float scale) {
    out[0] = acc[0] * scale;
}

// ---------------------------------------------------------------------------
extern "C" void kernel_launch(void* const* d_in, const int* in_sizes, int n_in,
                              void* d_out, int out_size, void* d_ws, size_t ws_size,
                              hipStream_t stream) {
    (void)in_sizes; (void)n_in; (void)out_size;
    const float* inp = (const float*)d_in[0];
    const float* tgt = (const float*)d_in[1];
    float* out = (float*)d_out;

    // Geometry constants (double host math, cast to f32 like the reference).
    const double half_diag = (IMG / 2.0) * sqrt(2.0);
    const float  gmax    = (float)asin(half_diag / 1075.0);
    const float  step_gm = (2.0f * gmax) / (float)(N_DET - 1);
    const float  step_th = (float)(2.0 * 3.14159265358979323846 / (double)N_ANG);
    const double ray_len = IMG * sqrt(2.0);
    const float  dt      = (float)(ray_len / (double)(N_SAMP - 1));
    const float  t0      = (float)(1075.0 - ray_len / 2.0);
    const float  SCALE   = (float)(512.0 / 416.0 * 0.03);
    const float  final_k = dt * dt * SCALE / (float)N_RAYS;

    const size_t diff_bytes = (size_t)IMG * IMG * sizeof(float);

    if (ws_size >= diff_bytes + 16) {
        // Fast path: fold input-target into one diff image (linearity of the
        // projector) -> half the sampling work.
        float* Dimg = (float*)d_ws;
        float* acc  = (float*)((char*)d_ws + diff_bytes);
        fb_diff_kernel<<<(IMG * IMG + 255) / 256, 256, 0, stream>>>(inp, tgt, Dimg, acc);
        fb_ray_kernel<true><<<N_RAYS / 256, 256, 0, stream>>>(
            Dimg, nullptr, acc, gmax, step_gm, step_th, t0, dt);
        fb_finalize<<<1, 1, 0, stream>>>(acc, out, final_k);
    } else {
        // Fallback: sample both images directly.
        float* acc = (float*)d_ws;
        fb_zero_acc<<<1, 1, 0, stream>>>(acc);
        fb_ray_kernel<false><<<N_RAYS / 256, 256, 0, stream>>>(
            inp, tgt, acc, gmax, step_gm, step_th, t0, dt);
        fb_finalize<<<1, 1, 0, stream>>>(acc, out, final_k);
    }
}